// GCN_Classifier_34815004901849
// MI455X (gfx1250) — compile-verified
//
#include <hip/hip_runtime.h>
#include <math.h>

#define N_NODES 100000
#define N_EDGES 1200000
#define N_GRAPHS 2048
#define HID 64
#define EPS 1e-5f

typedef __attribute__((ext_vector_type(2))) float v2f;
typedef __attribute__((ext_vector_type(8))) float v8f;

// Native non-returning f32 atomic add (GLOBAL_ATOMIC_ADD_F32, device scope).
// Guarantees the hardware atomic path (no CAS loop) for the 230M scatter-adds
// that dominate this workload. Tracked by STOREcnt; s_endpgm waits idle.
__device__ __forceinline__ void atomAddF(float* p, float v) {
  asm volatile("global_atomic_add_f32 %0, %1, off scope:SCOPE_DEV"
               :: "v"(p), "v"(v) : "memory");
}

// ---------------------------------------------------------------------------
// GEMM: xw[N x 64] = x[N x K] @ w[K x 64], f32 via V_WMMA_F32_16X16X4_F32.
// One wave -> one 16-row strip x 64 cols (4 accumulators share the A frag).
// A 16x4 layout: lanes 0-15 hold K=0,1 (2 VGPRs), lanes 16-31 hold K=2,3.
// B 4x16 layout: lanes 0-15 hold N, K=0,1; lanes 16-31 hold K=2,3.
// C/D: VGPR r -> M=r (lanes 0-15) / M=8+r (lanes 16-31), N=lane&15.
// ---------------------------------------------------------------------------
__global__ __launch_bounds__(256) void k_gemm_xw(const float* __restrict__ x,
                                                 const float* __restrict__ w,
                                                 float* __restrict__ xw,
                                                 int N, int K) {
  int wave  = threadIdx.x >> 5;
  int lane  = threadIdx.x & 31;
  int strip = blockIdx.x * 8 + wave;          // 16-row strip index
  if (strip * 16 >= N) return;                // wave-uniform exit (EXEC stays full)
  int row0  = strip * 16;
  int m     = lane & 15;
  int khalf = (lane >> 4) * 2;                // 0 or 2

  v8f acc0 = {}; v8f acc1 = {}; v8f acc2 = {}; v8f acc3 = {};
  for (int kk = 0; kk < K; kk += 4) {
    const float* xp = x + (size_t)(row0 + m) * K + kk + khalf;
    v2f a; a.x = xp[0]; a.y = xp[1];

    const float* wp = w + (size_t)(kk + khalf) * HID + m;
    v2f b0, b1, b2, b3;
    b0.x = wp[0];  b0.y = wp[HID + 0];
    b1.x = wp[16]; b1.y = wp[HID + 16];
    b2.x = wp[32]; b2.y = wp[HID + 32];
    b3.x = wp[48]; b3.y = wp[HID + 48];

    acc0 = __builtin_amdgcn_wmma_f32_16x16x4_f32(false, a, false, b0, (short)0, acc0, false, false);
    acc1 = __builtin_amdgcn_wmma_f32_16x16x4_f32(false, a, false, b1, (short)0, acc1, false, false);
    acc2 = __builtin_amdgcn_wmma_f32_16x16x4_f32(false, a, false, b2, (short)0, acc2, false, false);
    acc3 = __builtin_amdgcn_wmma_f32_16x16x4_f32(false, a, false, b3, (short)0, acc3, false, false);
  }

  int n = lane & 15;
  int mbase = (lane < 16) ? 0 : 8;
#pragma unroll
  for (int r = 0; r < 8; ++r) {
    float* op = xw + (size_t)(row0 + mbase + r) * HID + n;
    op[0]  = acc0[r];
    op[16] = acc1[r];
    op[32] = acc2[r];
    op[48] = acc3[r];
  }
}

// ---------------------------------------------------------------------------
// Degree / normalization
// ---------------------------------------------------------------------------
__global__ void k_fill_int(int* p, int v, int n) {
  int i = blockIdx.x * blockDim.x + threadIdx.x;
  if (i < n) p[i] = v;
}
__global__ void k_zero_f(float* p, int n) {
  int i = blockIdx.x * blockDim.x + threadIdx.x;
  if (i < n) p[i] = 0.0f;
}
__global__ void k_deg(const int* __restrict__ dst, int* __restrict__ deg, int E) {
  int e = blockIdx.x * blockDim.x + threadIdx.x;
  if (e < E) atomicAdd(&deg[dst[e]], 1);
}
__global__ void k_dinv(const int* __restrict__ deg, float* __restrict__ dinv, int N) {
  int i = blockIdx.x * blockDim.x + threadIdx.x;
  if (i < N) dinv[i] = rsqrtf((float)deg[i]);   // deg >= 1 (self-loop)
}

// Self-loop contribution initializes the aggregation buffer (full overwrite).
__global__ void k_selfloop(const float* __restrict__ xw, const float* __restrict__ dinv,
                           float* __restrict__ y, int N) {
  size_t idx = (size_t)blockIdx.x * blockDim.x + threadIdx.x;  // over N*64
  if (idx >= (size_t)N * HID) return;
  int i = (int)(idx >> 6);
  float di = dinv[i];
  y[idx] = xw[idx] * di * di;
}

// Edge scatter: 64 threads per edge (coalesced 256B gather, uniform s/d per group).
__global__ __launch_bounds__(256) void k_edges(const float* __restrict__ xw,
                                               const int* __restrict__ src,
                                               const int* __restrict__ dst,
                                               const float* __restrict__ dinv,
                                               float* __restrict__ y, int E) {
  int e = blockIdx.x * 4 + (threadIdx.x >> 6);
  int c = threadIdx.x & 63;
  if (e >= E) return;
  int s = src[e], d = dst[e];
  float norm = dinv[s] * dinv[d];
  atomAddF(&y[(size_t)d * HID + c], xw[(size_t)s * HID + c] * norm);
}

// ---------------------------------------------------------------------------
// BatchNorm (training-mode stats) + ReLU.  GCN bias cancels through BN.
// ---------------------------------------------------------------------------
__global__ void k_stats(const float* __restrict__ y, float* __restrict__ sums, int N) {
  int c  = threadIdx.x & 63;
  int rg = threadIdx.x >> 6;                 // 0..3 row groups per block
  float s = 0.0f, s2 = 0.0f;
  for (int i = blockIdx.x * 4 + rg; i < N; i += gridDim.x * 4) {
    float v = y[(size_t)i * HID + c];
    s += v; s2 += v * v;
  }
  atomAddF(&sums[c], s);
  atomAddF(&sums[HID + c], s2);
}

__global__ void k_bn_relu(float* __restrict__ y, const float* __restrict__ sums,
                          const float* __restrict__ gam, const float* __restrict__ bet,
                          int N) {
  size_t idx = (size_t)blockIdx.x * blockDim.x + threadIdx.x;  // over N*64
  if (idx >= (size_t)N * HID) return;
  int c = (int)(idx & 63);
  float invN = 1.0f / (float)N;
  float mu  = sums[c] * invN;
  float var = sums[HID + c] * invN - mu * mu;
  float v = (y[idx] - mu) * rsqrtf(var + EPS) * gam[c] + bet[c];
  y[idx] = v > 0.0f ? v : 0.0f;
}

// ---------------------------------------------------------------------------
// Pooling: per-graph mean (atomic f32 add) + max (int-bit atomicMax, vals >= 0)
// ---------------------------------------------------------------------------
__global__ void k_count(const int* __restrict__ batch, int* __restrict__ cnts, int N) {
  int i = blockIdx.x * blockDim.x + threadIdx.x;
  if (i < N) atomicAdd(&cnts[batch[i]], 1);
}
__global__ void k_pool(const float* __restrict__ x, const int* __restrict__ batch,
                       float* __restrict__ meanAcc, float* __restrict__ maxBuf, int N) {
  size_t idx = (size_t)blockIdx.x * blockDim.x + threadIdx.x;  // over N*64
  if (idx >= (size_t)N * HID) return;
  int i = (int)(idx >> 6);
  int c = (int)(idx & 63);
  int g = batch[i];
  float v = x[idx];
  atomAddF(&meanAcc[(size_t)g * HID + c], v);
  atomicMax((int*)&maxBuf[(size_t)g * HID + c], __float_as_int(v));  // v >= 0
}

// ---------------------------------------------------------------------------
// Classifier MLP (tiny): 138->64 relu, 64->32 relu, 32->1 sigmoid
// ---------------------------------------------------------------------------
__global__ void k_mlp1(const float* __restrict__ meanAcc, const int* __restrict__ cnts,
                       const float* __restrict__ maxBuf, const float* __restrict__ u,
                       const float* __restrict__ fw1, const float* __restrict__ fb1,
                       float* __restrict__ h1) {
  int idx = blockIdx.x * blockDim.x + threadIdx.x;   // over G*64
  if (idx >= N_GRAPHS * HID) return;
  int g = idx >> 6, j = idx & 63;
  int cnt = cnts[g]; if (cnt < 1) cnt = 1;
  float inv = 1.0f / (float)cnt;
  float acc = fb1[j];
  for (int k = 0; k < HID; ++k) acc += meanAcc[g * HID + k] * inv * fw1[k * HID + j];
  for (int k = 0; k < HID; ++k) acc += maxBuf[g * HID + k] * fw1[(HID + k) * HID + j];
  for (int k = 0; k < 10;  ++k) acc += u[g * 10 + k] * fw1[(2 * HID + k) * HID + j];
  h1[idx] = acc > 0.0f ? acc : 0.0f;
}
__global__ void k_mlp2(const float* __restrict__ h1, const float* __restrict__ fw2,
                       const float* __restrict__ fb2, float* __restrict__ h2) {
  int idx = blockIdx.x * blockDim.x + threadIdx.x;   // over G*32
  if (idx >= N_GRAPHS * 32) return;
  int g = idx >> 5, j = idx & 31;
  float acc = fb2[j];
  for (int k = 0; k < HID; ++k) acc += h1[g * HID + k] * fw2[k * 32 + j];
  h2[idx] = acc > 0.0f ? acc : 0.0f;
}
__global__ void k_mlp3(const float* __restrict__ h2, const float* __restrict__ fw3,
                       const float* __restrict__ fb3, float* __restrict__ out) {
  int g = blockIdx.x * blockDim.x + threadIdx.x;
  if (g >= N_GRAPHS) return;
  float acc = fb3[0];
  for (int k = 0; k < 32; ++k) acc += h2[g * 32 + k] * fw3[k];
  out[g] = 1.0f / (1.0f + expf(-acc));
}

// ---------------------------------------------------------------------------
extern "C" void kernel_launch(void* const* d_in, const int* in_sizes, int n_in,
                              void* d_out, int out_size, void* d_ws, size_t ws_size,
                              hipStream_t stream) {
  const float* x     = (const float*)d_in[0];
  const int*   eidx  = (const int*)  d_in[1];
  const int*   batch = (const int*)  d_in[2];
  const float* u     = (const float*)d_in[3];
  const float* w[3]   = {(const float*)d_in[4],  (const float*)d_in[6],  (const float*)d_in[8]};
  const float* gam[3] = {(const float*)d_in[10], (const float*)d_in[12], (const float*)d_in[14]};
  const float* bet[3] = {(const float*)d_in[11], (const float*)d_in[13], (const float*)d_in[15]};
  const float* fw1 = (const float*)d_in[16]; const float* fb1 = (const float*)d_in[17];
  const float* fw2 = (const float*)d_in[18]; const float* fb2 = (const float*)d_in[19];
  const float* fw3 = (const float*)d_in[20]; const float* fb3 = (const float*)d_in[21];
  float* out = (float*)d_out;

  const int* src = eidx;
  const int* dst = eidx + N_EDGES;

  // Workspace carve-up
  char* ws = (char*)d_ws;
  const size_t NF = (size_t)N_NODES * HID;
  float* buf0    = (float*)ws;                       ws += NF * 4;                 // features / aggregation
  float* buf1    = (float*)ws;                       ws += NF * 4;                 // x @ W
  int*   degi    = (int*)ws;                         ws += (size_t)N_NODES * 4;
  float* dinv    = (float*)ws;                       ws += (size_t)N_NODES * 4;
  float* sums    = (float*)ws;                       ws += 2 * HID * 4;            // [sum|sumsq]
  float* meanAcc = (float*)ws;                       ws += (size_t)N_GRAPHS * HID * 4;
  float* maxBuf  = (float*)ws;                       ws += (size_t)N_GRAPHS * HID * 4;
  int*   cnts    = (int*)ws;                         ws += (size_t)N_GRAPHS * 4;
  float* h1      = (float*)ws;                       ws += (size_t)N_GRAPHS * HID * 4;
  float* h2      = (float*)ws;                       ws += (size_t)N_GRAPHS * 32 * 4;

  const int NE_BLK   = (N_NODES * HID + 255) / 256;      // elementwise over N*64
  const int STRIPS   = N_NODES / 16;                     // 6250, exact
  const int GEMM_BLK = (STRIPS + 7) / 8;

  // --- symmetric normalization ---
  k_fill_int<<<(N_NODES + 255) / 256, 256, 0, stream>>>(degi, 1, N_NODES);
  k_deg<<<(N_EDGES + 255) / 256, 256, 0, stream>>>(dst, degi, N_EDGES);
  k_dinv<<<(N_NODES + 255) / 256, 256, 0, stream>>>(degi, dinv, N_NODES);

  // --- 3x (GCNConv -> BN -> ReLU), ping-pong buf0/buf1 ---
  for (int L = 0; L < 3; ++L) {
    const float* xin = (L == 0) ? x : buf0;
    int K = (L == 0) ? 4 : HID;
    k_gemm_xw<<<GEMM_BLK, 256, 0, stream>>>(xin, w[L], buf1, N_NODES, K);
    k_selfloop<<<NE_BLK, 256, 0, stream>>>(buf1, dinv, buf0, N_NODES);
    k_edges<<<(N_EDGES + 3) / 4, 256, 0, stream>>>(buf1, src, dst, dinv, buf0, N_EDGES);
    k_zero_f<<<1, 2 * HID, 0, stream>>>(sums, 2 * HID);
    k_stats<<<512, 256, 0, stream>>>(buf0, sums, N_NODES);
    k_bn_relu<<<NE_BLK, 256, 0, stream>>>(buf0, sums, gam[L], bet[L], N_NODES);
  }

  // --- pooling ---
  k_zero_f<<<(2 * N_GRAPHS * HID + 255) / 256, 256, 0, stream>>>(meanAcc, 2 * N_GRAPHS * HID); // meanAcc+maxBuf contiguous
  k_fill_int<<<(N_GRAPHS + 255) / 256, 256, 0, stream>>>(cnts, 0, N_GRAPHS);
  k_count<<<(N_NODES + 255) / 256, 256, 0, stream>>>(batch, cnts, N_NODES);
  k_pool<<<NE_BLK, 256, 0, stream>>>(buf0, batch, meanAcc, maxBuf, N_NODES);

  // --- classifier ---
  k_mlp1<<<(N_GRAPHS * HID + 255) / 256, 256, 0, stream>>>(meanAcc, cnts, maxBuf, u, fw1, fb1, h1);
  k_mlp2<<<(N_GRAPHS * 32 + 255) / 256, 256, 0, stream>>>(h1, fw2, fb2, h2);
  k_mlp3<<<(N_GRAPHS + 255) / 256, 256, 0, stream>>>(h2, fw3, fb3, out);
}